// PartialCrossEntropyHead_59090160058442
// MI455X (gfx1250) — compile-verified
//
#include <hip/hip_runtime.h>

// CDNA5 / gfx1250 fused cross-entropy head:
//   evaluate_tokens = logits @ W^T   (fp32 GEMM via V_WMMA_F32_16X16X4_F32)
//   loss / token_accuracy / string_accuracy via fused row-softmax stats.

typedef float v2f __attribute__((ext_vector_type(2)));
typedef float v8f __attribute__((ext_vector_type(8)));

#define VOCAB   50257
#define DMODEL  1024
#define MROWS   4096      // BS * SEQ
#define SEQLEN  2048
#define BN      32        // vocab columns per block (N-strip)
#define BM      64        // token rows per tile iteration
#define BK      64        // K step staged in LDS for A
#define WSTRIDE 1028      // padded LDS row stride for W strip: (4n + k) % 64 conflict-free
#define ASTRIDE 68        // padded LDS row stride for A tile:  (4m + k) % 64 conflict-free
#define NSTRIPS ((VOCAB + BN - 1) / BN)   // 1571

// ---------------------------------------------------------------------------
// GEMM: each block owns a 32-wide vocab strip. The full W strip (32 x 1024
// fp32) is staged into LDS ONCE per block -> W is read from HBM exactly once
// across the whole kernel (206 MB, the roofline minimum). Logits (16.8 MB)
// stay L2-resident across strips. 8 waves/block, one 16x16 fp32 WMMA tile per
// wave, K accumulated with v_wmma_f32_16x16x4_f32.
// ---------------------------------------------------------------------------
__global__ __launch_bounds__(256) void ce_head_gemm(
    const float* __restrict__ logits,   // [4096, 1024]
    const float* __restrict__ W,        // [50257, 1024]
    float* __restrict__ evt)            // [4096, 50257]
{
  __shared__ float Wl[BN * WSTRIDE];    // 131584 B
  __shared__ float Al[BM * ASTRIDE];    // 17408 B  (total ~145.5 KB of 320 KB WGP LDS)

  const int tid   = threadIdx.x;
  const int lane  = tid & 31;           // wave32
  const int wv    = tid >> 5;           // 8 waves
  const int lm    = lane & 15;          // lane % 16
  const int lh    = lane >> 4;          // lane half (0/1)
  const int mloc  = (wv >> 1) * 16;     // wave's row-tile offset within BM
  const int nloc  = (wv & 1) * 16;      // wave's col-tile offset within BN
  const int nBase = blockIdx.x * BN;

  // ---- Stage the entire W strip (BN rows x DMODEL cols) into LDS once ----
  for (int c = tid; c < BN * (DMODEL / 4); c += 256) {
    int n  = c >> 8;                    // DMODEL/4 == 256 float4 per row
    int k4 = c & 255;
    int gn = nBase + n; if (gn >= VOCAB) gn = VOCAB - 1;   // clamp ragged edge
    float4 v = reinterpret_cast<const float4*>(W + (size_t)gn * DMODEL)[k4];
    reinterpret_cast<float4*>(Wl + n * WSTRIDE)[k4] = v;   // WSTRIDE%4==0 -> 16B aligned
  }
  __syncthreads();

  for (int m0 = 0; m0 < MROWS; m0 += BM) {
    v8f acc = {0.f, 0.f, 0.f, 0.f, 0.f, 0.f, 0.f, 0.f};

    for (int kb = 0; kb < DMODEL; kb += BK) {
      __syncthreads();                  // protect Al against previous readers
      // ---- Stage A tile (BM x BK) ----
      for (int c = tid; c < BM * (BK / 4); c += 256) {
        int r  = c >> 4;                // BK/4 == 16 float4 per row
        int k4 = c & 15;
        float4 v = reinterpret_cast<const float4*>(
            logits + (size_t)(m0 + r) * DMODEL + kb)[k4];
        reinterpret_cast<float4*>(Al + r * ASTRIDE)[k4] = v;
      }
      __syncthreads();

      // Fragment addressing per ISA 32-bit A layout: k = j + 2*(lane/16),
      // m/n = lane%16  -> each lane reads a contiguous float2.
      const float* ap = Al + (mloc + lm) * ASTRIDE + 2 * lh;
      const float* bp = Wl + (nloc + lm) * WSTRIDE + kb + 2 * lh;
      #pragma unroll
      for (int s = 0; s < BK; s += 4) {
        v2f a = *reinterpret_cast<const v2f*>(ap + s);
        v2f b = *reinterpret_cast<const v2f*>(bp + s);
        acc = __builtin_amdgcn_wmma_f32_16x16x4_f32(
            false, a, false, b, (short)0, acc, false, false);
      }
    }

    // ---- Write C tile: element (vgpr r, lane) = C[r + 8*lh][lm] ----
    int col = nBase + nloc + lm;
    if (col < VOCAB) {
      #pragma unroll
      for (int r = 0; r < 8; ++r) {
        int row = m0 + mloc + 8 * lh + r;
        evt[(size_t)row * VOCAB + col] = acc[r];
      }
    }
  }
}

// ---------------------------------------------------------------------------
// Per-row softmax stats + masked NLL / accuracy accumulation.
// One 256-thread block per token row; online (max, sumexp) merge + argmax
// (ties -> lowest index, matching jnp.argmax).
// accum layout (in d_ws): [0] nll_sum(f32) [1] n_valid(f32) [2] n_correct(f32)
//                         [3] seq0_all_correct(i32) [4] seq1_all_correct(i32)
// ---------------------------------------------------------------------------
__global__ __launch_bounds__(256) void ce_head_rowstats(
    const float* __restrict__ evt, const int* __restrict__ target,
    const int* __restrict__ mask, float* __restrict__ accum)
{
  const int row = blockIdx.x;
  const int tid = threadIdx.x;
  const float* x = evt + (size_t)row * VOCAB;

  float mx = -__builtin_inff(), sum = 0.f, bv = -__builtin_inff();
  int bi = 0x7fffffff;
  for (int c = tid; c < VOCAB; c += 256) {
    float v = x[c];
    if (v > mx) { sum *= __expf(mx - v); mx = v; }
    sum += __expf(v - mx);
    if (v > bv) { bv = v; bi = c; }     // strided ascending -> earliest kept
  }

  __shared__ float smx[256], ssum[256], sbv[256];
  __shared__ int   sbi[256];
  smx[tid] = mx; ssum[tid] = sum; sbv[tid] = bv; sbi[tid] = bi;
  __syncthreads();
  for (int off = 128; off; off >>= 1) {
    if (tid < off) {
      float m2 = smx[tid + off], s2 = ssum[tid + off];
      float M  = fmaxf(smx[tid], m2);
      ssum[tid] = ssum[tid] * __expf(smx[tid] - M) + s2 * __expf(m2 - M);
      smx[tid]  = M;
      float bv2 = sbv[tid + off]; int bi2 = sbi[tid + off];
      if (bv2 > sbv[tid] || (bv2 == sbv[tid] && bi2 < sbi[tid])) {
        sbv[tid] = bv2; sbi[tid] = bi2;
      }
    }
    __syncthreads();
  }

  if (tid == 0) {
    float logZ = smx[0] + __logf(ssum[0]);
    int   t    = target[row];
    float xt   = x[t];
    int   pred = sbi[0];
    bool  corr = (pred == t);
    if (mask[row] != 0) {
      atomicAdd(&accum[0], logZ - xt);            // nll = logZ - x_target
      atomicAdd(&accum[1], 1.0f);
      atomicAdd(&accum[2], corr ? 1.0f : 0.0f);
      if (!corr) atomicExch(((int*)accum) + 3 + (row / SEQLEN), 0);
    }
  }
}

__global__ void ce_head_init(float* accum) {
  accum[0] = 0.f; accum[1] = 0.f; accum[2] = 0.f;
  ((int*)accum)[3] = 1; ((int*)accum)[4] = 1;
}

__global__ void ce_head_finalize(const float* __restrict__ accum,
                                 float* __restrict__ out) {
  float nv = accum[1];
  const int* f = (const int*)accum;
  out[0] = accum[0] / nv;                              // loss
  out[1] = accum[2] / nv;                              // token_accuracy
  out[2] = 0.5f * ((float)f[3] + (float)f[4]);         // string_accuracy
}

extern "C" void kernel_launch(void* const* d_in, const int* in_sizes, int n_in,
                              void* d_out, int out_size, void* d_ws, size_t ws_size,
                              hipStream_t stream) {
  (void)in_sizes; (void)n_in; (void)out_size; (void)ws_size;
  const float* logits = (const float*)d_in[0];
  const int*   target = (const int*)d_in[1];
  const int*   mask   = (const int*)d_in[2];
  const float* W      = (const float*)d_in[3];
  float* out   = (float*)d_out;
  float* evt   = out + 3;                // [loss, tok_acc, str_acc, evaluate_tokens...]
  float* accum = (float*)d_ws;

  ce_head_init    <<<1,       1,   0, stream>>>(accum);
  ce_head_gemm    <<<NSTRIPS, 256, 0, stream>>>(logits, W, evt);
  ce_head_rowstats<<<MROWS,   256, 0, stream>>>(evt, target, mask, accum);
  ce_head_finalize<<<1,       1,   0, stream>>>(accum, out);
}